// FingerNetWrapper_52046413693087
// MI455X (gfx1250) — compile-verified
//
#include <hip/hip_runtime.h>
#include <hip/hip_bf16.h>
#include <stdint.h>

typedef float v2f __attribute__((ext_vector_type(2)));
typedef float v8f __attribute__((ext_vector_type(8)));

static constexpr int   NB   = 32;           // batch
static constexpr int   HW   = 128 * 128;    // coarse pixels per image
static constexpr int   KC   = 1024;         // top-k candidates
static constexpr int   FULL = 1024 * 1024;  // fine pixels per image
static constexpr float PI_F = 3.14159265358979f;
// gauss1d(ks=5, sigma=1.5), normalized
static constexpr float GW0 = 0.29208171f;
static constexpr float GW1 = 0.23388052f;
static constexpr float GW2 = 0.12007788f;

// ---------------- mask pipeline ----------------
__global__ __launch_bounds__(256) void k_hblur(const float* __restrict__ seg,
                                               float* __restrict__ hbuf) {
  int gid = blockIdx.x * 256 + threadIdx.x;
  if (gid >= NB * HW) return;
  int p = gid & (HW - 1);
  int col = p & 127;
  int base = gid - col;
  float s = 0.f;
#pragma unroll
  for (int dx = -2; dx <= 2; ++dx) {
    int c = col + dx;
    if (c < 0 || c > 127) continue;
    float w = (dx == 0) ? GW0 : ((dx == 1 || dx == -1) ? GW1 : GW2);
    s += w * rintf(seg[base + c]);  // rintf == round-half-even == jnp.round
  }
  hbuf[gid] = s;
}

__global__ __launch_bounds__(256) void k_vblur(const float* __restrict__ hbuf,
                                               float* __restrict__ mask) {
  int gid = blockIdx.x * 256 + threadIdx.x;
  if (gid >= NB * HW) return;
  int p = gid & (HW - 1);
  int row = p >> 7;
  float s = 0.f;
#pragma unroll
  for (int dy = -2; dy <= 2; ++dy) {
    int r = row + dy;
    if (r < 0 || r > 127) continue;
    float w = (dy == 0) ? GW0 : ((dy == 1 || dy == -1) ? GW1 : GW2);
    s += w * hbuf[gid + dy * 128];
  }
  mask[gid] = rintf(s);
}

// ---------------- masked score + sort keys ----------------
__global__ __launch_bounds__(256) void k_score_key(const float* __restrict__ mscore,
                                                   const float* __restrict__ mask,
                                                   float* __restrict__ msout,
                                                   unsigned long long* __restrict__ keys) {
  int gid = blockIdx.x * 256 + threadIdx.x;
  if (gid >= NB * HW) return;
  int p = gid & (HW - 1);
  float s = mscore[gid] * mask[gid];
  msout[gid] = s;
  unsigned u = __float_as_uint(s);
  u = (u & 0x80000000u) ? ~u : (u | 0x80000000u);  // order-preserving map
  // descending sort, smaller index wins ties -> larger low word for smaller idx
  keys[gid] = ((unsigned long long)u << 32) | (unsigned)(16383 - p);
}

// ---------------- LDS-resident bitonic sort (top-k) ----------------
// One block per batch: 16384 u64 keys (128 KB) live entirely in LDS
// (gfx1250 WGP: 320 KB LDS per workgroup). 105 (k,j) passes, 8 disjoint
// compare-exchanges per thread per pass, 1 barrier per pass.
__global__ __launch_bounds__(1024) void k_sort(const unsigned long long* __restrict__ keys_in,
                                               unsigned long long* __restrict__ topk) {
  extern __shared__ unsigned long long sk[];   // 16384 entries
  int b = blockIdx.x, tid = threadIdx.x;
  const unsigned long long* kb = keys_in + (size_t)b * HW;
#pragma unroll
  for (int w = 0; w < 16; ++w) sk[w * 1024 + tid] = kb[w * 1024 + tid];
  __syncthreads();
  for (int k = 2; k <= HW; k <<= 1) {
    for (int j = k >> 1; j > 0; j >>= 1) {
#pragma unroll
      for (int w = 0; w < 8; ++w) {
        int p = w * 1024 + tid;                       // pair id 0..8191
        int i = ((p & ~(j - 1)) << 1) | (p & (j - 1));
        int ixj = i + j;
        unsigned long long a = sk[i], c = sk[ixj];
        bool up = (i & k) == 0;                       // descending overall
        if (up ? (a < c) : (a > c)) { sk[i] = c; sk[ixj] = a; }
      }
      __syncthreads();
    }
  }
  topk[(size_t)b * KC + tid] = sk[tid];               // sorted top-1024
}

// ---------------- per-candidate argmaxes ----------------
__global__ __launch_bounds__(256) void k_cand(const unsigned long long* __restrict__ topk,
                                              const float* __restrict__ msout,
                                              const float* __restrict__ morient,
                                              const float* __restrict__ mxoff,
                                              const float* __restrict__ myoff,
                                              float* __restrict__ Xc, float* __restrict__ Yc,
                                              float* __restrict__ ANGc, float* __restrict__ SCc,
                                              float* __restrict__ SSc, float* __restrict__ VALc,
                                              float* __restrict__ out_min) {
  int gid = blockIdx.x * 256 + threadIdx.x;
  if (gid >= NB * KC) return;
  int b = gid >> 10;
  unsigned long long key = topk[gid];
  int idx = 16383 - (int)(unsigned)(key & 0xffffffffull);
  float score = msout[b * HW + idx];
  int row = idx >> 7, col = idx & 127;

  const float* ob = morient + (size_t)b * 90 * HW + idx;
  int ba = 0; float bv = ob[0];
  for (int c = 1; c < 90; ++c) { float v = ob[(size_t)c * HW]; if (v > bv) { bv = v; ba = c; } }

  const float* xb = mxoff + (size_t)b * 8 * HW + idx;
  int bx = 0; float bxv = xb[0];
#pragma unroll
  for (int c = 1; c < 8; ++c) { float v = xb[(size_t)c * HW]; if (v > bxv) { bxv = v; bx = c; } }

  const float* yb = myoff + (size_t)b * 8 * HW + idx;
  int by = 0; float byv = yb[0];
#pragma unroll
  for (int c = 1; c < 8; ++c) { float v = yb[(size_t)c * HW]; if (v > byv) { byv = v; by = c; } }

  float ang = ((float)ba * 2.f - 89.f) * (PI_F / 180.f);
  float x = (float)col * 8.f + (float)bx;
  float y = (float)row * 8.f + (float)by;

  Xc[gid] = x; Yc[gid] = y; ANGc[gid] = ang; SCc[gid] = score;
  SSc[gid] = x * x + y * y;
  VALc[gid] = (score > 0.5f) ? 1.f : 0.f;
  float* o = out_min + (size_t)gid * 5;
  o[0] = x; o[1] = y; o[2] = ang; o[3] = score;
}

// ---------------- WMMA suppression matrix ----------------
// dist^2(i,j) via K=4 fp32 WMMA:  A_i=[x_i, y_i, s_i, 1], B_j=[-2x_j, -2y_j, 1, s_j]
// One wave computes a 16x16 tile; ballot packs (dist<16 && angdiff<pi/6) into bits.
__global__ __launch_bounds__(256) void k_sup(const float* __restrict__ Xc,
                                             const float* __restrict__ Yc,
                                             const float* __restrict__ ANGc,
                                             const float* __restrict__ SSc,
                                             unsigned short* __restrict__ sup) {
  int wave = (blockIdx.x * 256 + threadIdx.x) >> 5;  // 131072 waves total
  int lane = threadIdx.x & 31;
  int b  = wave >> 12;
  int rm = wave & 4095;
  int ti = rm >> 6, tj = rm & 63;

  const float* Xb = Xc + b * KC;  const float* Yb = Yc + b * KC;
  const float* Ab = ANGc + b * KC; const float* Sb = SSc + b * KC;

  int l = lane & 15;
  int i0 = ti * 16 + l, j0 = tj * 16 + l;
  v2f av, bv;
  if (lane < 16) {            // K = 0,1
    av.x = Xb[i0];         av.y = Yb[i0];
    bv.x = -2.f * Xb[j0];  bv.y = -2.f * Yb[j0];
  } else {                    // K = 2,3
    av.x = Sb[i0];         av.y = 1.f;
    bv.x = 1.f;            bv.y = Sb[j0];
  }
  v8f c = {};
  c = __builtin_amdgcn_wmma_f32_16x16x4_f32(false, av, false, bv,
                                            (short)0, c, false, false);
  float aj = Ab[j0];  // column angle for this lane
#pragma unroll
  for (int v = 0; v < 8; ++v) {
    int rowi = ti * 16 + v + ((lane >> 4) << 3);   // C layout: lanes16-31 hold M=v+8
    float ai = Ab[rowi];
    float da = fabsf(ai - aj);
    float angd = fminf(da, 2.f * PI_F - da);
    bool pred = (c[v] < 256.0f) && (angd < (PI_F / 6.f));
    unsigned m = (unsigned)__ballot(pred);
    if (lane == 0) {
      sup[((size_t)b * KC + ti * 16 + v) * 64 + tj]     = (unsigned short)(m & 0xFFFFu);
      sup[((size_t)b * KC + ti * 16 + v + 8) * 64 + tj] = (unsigned short)(m >> 16);
    }
  }
}

// ---------------- sequential NMS keep loop ----------------
__global__ __launch_bounds__(1024) void k_nms(const unsigned short* __restrict__ sup,
                                              const float* __restrict__ VALc,
                                              float* __restrict__ out_min) {
  int b = blockIdx.x;
  int j = threadIdx.x;
  __shared__ unsigned char keep[KC];
  keep[j] = VALc[b * KC + j] > 0.5f ? 1 : 0;
  __syncthreads();
  const unsigned short* sb = sup + (size_t)b * KC * 64;
  for (int i = 0; i < KC - 1; ++i) {
    if (keep[i] && j > i) {
      unsigned short w = sb[(size_t)i * 64 + (j >> 4)];
      if ((w >> (j & 15)) & 1) keep[j] = 0;
    }
    __syncthreads();
  }
  out_min[((size_t)b * KC + j) * 5 + 4] = (float)keep[j];
}

// ---------------- full-image orientation argmax ----------------
__global__ __launch_bounds__(256) void k_oriarg(const float* __restrict__ orient,
                                                float* __restrict__ orival) {
  int gid = blockIdx.x * 256 + threadIdx.x;
  if (gid >= NB * HW) return;
  int b = gid >> 14, p = gid & (HW - 1);
  const float* ob = orient + (size_t)b * 90 * HW + p;
  int ba = 0; float bv = ob[0];
  for (int c = 1; c < 90; ++c) { float v = ob[(size_t)c * HW]; if (v > bv) { bv = v; ba = c; } }
  orival[gid] = ((float)ba * 2.f - 89.f) * (PI_F / 180.f);
}

// ---------------- per-batch min/max of masked enhanced image (float4) ----------------
__global__ __launch_bounds__(256) void k_minmax1(const float* __restrict__ enh,
                                                 const float* __restrict__ mask,
                                                 float* __restrict__ pmin,
                                                 float* __restrict__ pmax) {
  int b = blockIdx.y, blk = blockIdx.x, tid = threadIdx.x;
  __shared__ float smin[256], smax[256];
  float mn = 3.4e38f, mx = -3.4e38f;
  const float4* eb = (const float4*)(enh + (size_t)b * FULL);
  int base4 = blk * (FULL / 64 / 4);                 // 4096 float4 per block
  for (int p4 = base4 + tid; p4 < base4 + FULL / 64 / 4; p4 += 256) {
    int p = p4 << 2;
    int y = p >> 10, x = p & 1023;                   // 4-group stays in one coarse cell
    float cm = mask[b * HW + (y >> 3) * 128 + (x >> 3)];
    float4 e = eb[p4];
    mn = fminf(mn, fminf(fminf(e.x * cm, e.y * cm), fminf(e.z * cm, e.w * cm)));
    mx = fmaxf(mx, fmaxf(fmaxf(e.x * cm, e.y * cm), fmaxf(e.z * cm, e.w * cm)));
  }
  smin[tid] = mn; smax[tid] = mx;
  __syncthreads();
  for (int s = 128; s > 0; s >>= 1) {
    if (tid < s) { smin[tid] = fminf(smin[tid], smin[tid + s]);
                   smax[tid] = fmaxf(smax[tid], smax[tid + s]); }
    __syncthreads();
  }
  if (tid == 0) { pmin[b * 64 + blk] = smin[0]; pmax[b * 64 + blk] = smax[0]; }
}

__global__ __launch_bounds__(64) void k_minmax2(const float* __restrict__ pmin,
                                                const float* __restrict__ pmax,
                                                float* __restrict__ mm) {
  int b = blockIdx.x, tid = threadIdx.x;
  __shared__ float smin[64], smax[64];
  smin[tid] = pmin[b * 64 + tid];
  smax[tid] = pmax[b * 64 + tid];
  __syncthreads();
  for (int s = 32; s > 0; s >>= 1) {
    if (tid < s) { smin[tid] = fminf(smin[tid], smin[tid + s]);
                   smax[tid] = fmaxf(smax[tid], smax[tid + s]); }
    __syncthreads();
  }
  if (tid == 0) { mm[2 * b] = smin[0]; mm[2 * b + 1] = smax[0]; }
}

// ---------------- fused final elementwise pass (float4, 3 image outputs) ----------------
__global__ __launch_bounds__(256) void k_final(const float* __restrict__ enh,
                                               const float* __restrict__ mask,
                                               const float* __restrict__ orival,
                                               const float* __restrict__ mm,
                                               float* __restrict__ out_enh,
                                               float* __restrict__ out_seg,
                                               float* __restrict__ out_of) {
  int gid = blockIdx.x * 256 + threadIdx.x;          // one float4 per thread
  if (gid >= NB * FULL / 4) return;
  int q = gid << 2;
  int b = q >> 20, pos = q & (FULL - 1);
  int y = pos >> 10, x = pos & 1023;
  int coarse = b * HW + (y >> 3) * 128 + (x >> 3);   // shared by the 4-group
  float cm = mask[coarse];
  float ofv = orival[coarse] * cm;
  float segv = (float)(unsigned char)(cm * 255.0f);
  float mn = mm[2 * b], mx = mm[2 * b + 1];
  float den = mx - mn + 1e-8f;

  float4 e4 = ((const float4*)enh)[gid];
  float4 oe;
  oe.x = (float)(unsigned char)((e4.x * cm - mn) / den * 255.0f);
  oe.y = (float)(unsigned char)((e4.y * cm - mn) / den * 255.0f);
  oe.z = (float)(unsigned char)((e4.z * cm - mn) / den * 255.0f);
  oe.w = (float)(unsigned char)((e4.w * cm - mn) / den * 255.0f);
  ((float4*)out_enh)[gid] = oe;
  ((float4*)out_seg)[gid] = make_float4(segv, segv, segv, segv);
  ((float4*)out_of)[gid]  = make_float4(ofv, ofv, ofv, ofv);
}

extern "C" void kernel_launch(void* const* d_in, const int* in_sizes, int n_in,
                              void* d_out, int out_size, void* d_ws, size_t ws_size,
                              hipStream_t stream) {
  const float* seg     = (const float*)d_in[0];  // (32,1,128,128)
  const float* mscore  = (const float*)d_in[1];  // (32,1,128,128)
  const float* morient = (const float*)d_in[2];  // (32,90,128,128)
  const float* mxoff   = (const float*)d_in[3];  // (32,8,128,128)
  const float* myoff   = (const float*)d_in[4];  // (32,8,128,128)
  const float* orient  = (const float*)d_in[5];  // (32,90,128,128)
  const float* enh     = (const float*)d_in[6];  // (32,1,1024,1024)

  float* out = (float*)d_out;
  float* out_min = out;                                   // 32*1024*5
  float* out_enh = out + (size_t)NB * KC * 5;             // 32*1024*1024
  float* out_seg = out_enh + (size_t)NB * FULL;
  float* out_of  = out_seg + (size_t)NB * FULL;

  // workspace layout (float offsets)
  float* ws = (float*)d_ws;
  float* hbuf   = ws;                         // 524288
  float* mask   = ws + 524288;                // 524288
  float* msout  = ws + 1048576;               // 524288
  float* orival = ws + 1572864;               // 524288
  unsigned long long* keys = (unsigned long long*)(ws + 2097152);  // 524288 u64
  unsigned long long* topk = (unsigned long long*)(ws + 3145728);  // 32768 u64
  float* Xc   = ws + 3211264;                 // 32768 each
  float* Yc   = Xc + 32768;
  float* ANGc = Yc + 32768;
  float* SCc  = ANGc + 32768;
  float* SSc  = SCc + 32768;
  float* VALc = SSc + 32768;
  unsigned short* sup = (unsigned short*)(ws + 3407872);  // 32*1024*64 u16 (4MB)
  float* pmin = ws + 3407872 + 1048576;       // 2048
  float* pmax = pmin + 2048;                  // 2048
  float* mm   = pmax + 2048;                  // 64

  const int gHW = (NB * HW + 255) / 256;      // 2048 blocks

  k_hblur<<<gHW, 256, 0, stream>>>(seg, hbuf);
  k_vblur<<<gHW, 256, 0, stream>>>(hbuf, mask);
  k_score_key<<<gHW, 256, 0, stream>>>(mscore, mask, msout, keys);

  // per-batch LDS-resident bitonic sort: 128 KB dynamic LDS (WGP has 320 KB)
  k_sort<<<NB, 1024, HW * sizeof(unsigned long long), stream>>>(keys, topk);

  k_cand<<<(NB * KC + 255) / 256, 256, 0, stream>>>(topk, msout, morient, mxoff, myoff,
                                                    Xc, Yc, ANGc, SCc, SSc, VALc, out_min);
  // 32 batches * 64*64 tiles, 1 tile per wave, 8 waves per block
  k_sup<<<16384, 256, 0, stream>>>(Xc, Yc, ANGc, SSc, sup);
  k_nms<<<NB, 1024, 0, stream>>>(sup, VALc, out_min);

  k_oriarg<<<gHW, 256, 0, stream>>>(orient, orival);
  k_minmax1<<<dim3(64, NB), 256, 0, stream>>>(enh, mask, pmin, pmax);
  k_minmax2<<<NB, 64, 0, stream>>>(pmin, pmax, mm);
  k_final<<<(NB * FULL / 4 + 255) / 256, 256, 0, stream>>>(enh, mask, orival, mm,
                                                           out_enh, out_seg, out_of);
}